// NeuralNDCGLoss_56006373540079
// MI455X (gfx1250) — compile-verified
//
#include <hip/hip_runtime.h>
#include <math.h>

// NeuralNDCG loss for MI455X (gfx1250, wave32).
// One workgroup (256 threads = 8 waves) per batch row; the whole 256x256
// matrix lives in LDS (stride-padded to 257 floats). Sinkhorn is run in the
// implicit diag(r)*P0*diag(c) form, so each iteration is just two
// WMMA-based weighted reductions (V_WMMA_F32_16X16X4_F32) + 256-elem vector
// updates. No HBM traffic inside the 50-iteration loop.

typedef float v2f __attribute__((ext_vector_type(2)));
typedef float v8f __attribute__((ext_vector_type(8)));

#define NN 256
#define SROW 257          // padded LDS row stride (bank-conflict-free both ways)
#define EPSV 1e-10f
#define MAXIT 50

__device__ __forceinline__ float wave_max(float v) {
  #pragma unroll
  for (int off = 16; off > 0; off >>= 1) v = fmaxf(v, __shfl_xor(v, off, 32));
  return v;
}
__device__ __forceinline__ float wave_sum(float v) {
  #pragma unroll
  for (int off = 16; off > 0; off >>= 1) v += __shfl_xor(v, off, 32);
  return v;
}

#if __has_builtin(__builtin_amdgcn_wmma_f32_16x16x4_f32)
#define HAVE_WMMA_F32 1
#else
#define HAVE_WMMA_F32 0
#endif

// rowT[i0+m] (m=0..15) = sum_j P[i0+m][j] * w[j], via 64 chained 16x16x4 WMMAs.
__device__ __forceinline__ void row_combine(const float* __restrict__ P,
                                            const float* __restrict__ w,
                                            float* __restrict__ outT,
                                            int i0, int lane) {
#if HAVE_WMMA_F32
  const int kb = (lane < 16) ? 0 : 2;
  const int m  = i0 + (lane & 15);
  v8f acc = {0.f,0.f,0.f,0.f,0.f,0.f,0.f,0.f};
  for (int j0 = 0; j0 < NN; j0 += 4) {
    v2f a, b;
    a.x = P[m * SROW + j0 + kb];
    a.y = P[m * SROW + j0 + kb + 1];
    b.x = w[j0 + kb];
    b.y = w[j0 + kb + 1];
    acc = __builtin_amdgcn_wmma_f32_16x16x4_f32(false, a, false, b,
                                                (short)0, acc, false, false);
  }
  if (lane == 0 || lane == 16) {
    const int base = i0 + ((lane == 16) ? 8 : 0);
    #pragma unroll
    for (int v = 0; v < 8; ++v) outT[base + v] = acc[v];
  }
#else
  // fallback (host pass / builtin missing): lanes 0..15 each do one row
  if (lane < 16) {
    const int m = i0 + lane;
    float s = 0.f;
    for (int j = 0; j < NN; ++j) s += P[m * SROW + j] * w[j];
    outT[m] = s;
  }
#endif
}

// outS[j0+n] (n=0..15) = sum_i r[i] * P[i][j0+n], via 64 chained WMMAs.
__device__ __forceinline__ void col_combine(const float* __restrict__ P,
                                            const float* __restrict__ r,
                                            float* __restrict__ outS,
                                            int j0, int lane) {
#if HAVE_WMMA_F32
  const int kb = (lane < 16) ? 0 : 2;
  const int n  = lane & 15;
  v8f acc = {0.f,0.f,0.f,0.f,0.f,0.f,0.f,0.f};
  for (int i0 = 0; i0 < NN; i0 += 4) {
    v2f a, b;
    a.x = r[i0 + kb];
    a.y = r[i0 + kb + 1];
    b.x = P[(i0 + kb)     * SROW + j0 + n];
    b.y = P[(i0 + kb + 1) * SROW + j0 + n];
    acc = __builtin_amdgcn_wmma_f32_16x16x4_f32(false, a, false, b,
                                                (short)0, acc, false, false);
  }
  if (lane < 16) outS[j0 + n] = acc[0];
#else
  if (lane < 16) {
    const int j = j0 + lane;
    float s = 0.f;
    for (int i = 0; i < NN; ++i) s += r[i] * P[i * SROW + j];
    outS[j] = s;
  }
#endif
}

extern "C" __global__ __launch_bounds__(256, 1)
void neural_ndcg_main(const float* __restrict__ y_pred,
                      const float* __restrict__ y_true,
                      float* __restrict__ ndcg_out,
                      float* __restrict__ flag_out) {
  extern __shared__ float smem[];
  float* P    = smem;                 // 256 * 257
  float* smv  = P    + NN * SROW;     // masked scores for A_s
  float* s0   = smv  + NN;            // scores with mask->0
  float* gn   = s0   + NN;            // gains 2^rel - 1 (mask->0)
  float* Ar   = gn   + NN;            // row sums of |s_i - s_k|
  float* rV   = Ar   + NN;            // Sinkhorn row scale
  float* cV   = rV   + NN;            // Sinkhorn col scale
  float* colS = cV   + NN;
  float* rowT = colS + NN;
  float* yt   = rowT + NN;
  float* red  = yt   + NN;
  float* w2   = red  + NN;
  __shared__ int s_nmask;

  const int b    = blockIdx.x;
  const int tid  = threadIdx.x;
  const int lane = tid & 31;
  const int wv   = tid >> 5;
  const float NEG_INF = -__builtin_inff();

  // ---- phase 0: load vectors, build per-element arrays ----
  {
    const float yp  = y_pred[b * NN + tid];
    const float ytv = y_true[b * NN + tid];
    const bool  mk  = (ytv == -1.0f);
    yt[tid]  = ytv;
    smv[tid] = mk ? -1e8f : yp;
    s0[tid]  = mk ? 0.0f  : yp;
    gn[tid]  = mk ? 0.0f  : (exp2f(ytv) - 1.0f);
    if (tid == 0) s_nmask = 0;
  }
  __syncthreads();
  {
    const bool mk = (yt[tid] == -1.0f);
    if (mk) atomicAdd(&s_nmask, 1);
  }
  // A_s row sums: Ar[j] = sum_k |smv[j] - smv[k]|
  {
    const float sj = smv[tid];
    float acc = 0.f;
    for (int k = 0; k < NN; ++k) acc += fabsf(sj - smv[k]);
    Ar[tid] = acc;
  }
  __syncthreads();
  const float v_valid = (float)(NN - s_nmask);

  // ---- phase 1: P0 = softmax rows of P_max with pad overrides folded in ----
  // P_max[i][j] = s0[j]*scal(i) - Ar[j];   pad_and -> 1, pad_or -> -inf
  // then Sinkhorn pre-mask: pad_or -> 0, pad_and -> 1 (folded into the store)
  for (int i = wv * 32; i < wv * 32 + 32; ++i) {
    const bool  mi   = (yt[i] == -1.0f);
    const float scal = ((float)i < v_valid) ? (v_valid - 1.0f - 2.0f * (float)i) : 0.0f;
    if (mi) {
      #pragma unroll
      for (int t = 0; t < 8; ++t) {
        const int j = lane + 32 * t;
        P[i * SROW + j] = (yt[j] == -1.0f) ? 1.0f : 0.0f;
      }
    } else {
      float vals[8];
      float mx = NEG_INF;
      #pragma unroll
      for (int t = 0; t < 8; ++t) {
        const int j  = lane + 32 * t;
        const bool mj = (yt[j] == -1.0f);
        const float vv = mj ? NEG_INF : (s0[j] * scal - Ar[j]);
        vals[t] = vv;
        mx = fmaxf(mx, vv);
      }
      mx = wave_max(mx);
      float ssum = 0.f;
      #pragma unroll
      for (int t = 0; t < 8; ++t) {
        const float e = (vals[t] == NEG_INF) ? 0.f : expf(vals[t] - mx);
        vals[t] = e;
        ssum += e;
      }
      ssum = wave_sum(ssum);
      #pragma unroll
      for (int t = 0; t < 8; ++t) {
        const int j = lane + 32 * t;
        P[i * SROW + j] = vals[t] / ssum;
      }
    }
  }
  rV[tid] = 1.0f;
  cV[tid] = 1.0f;
  __syncthreads();

  // ---- phase 2: Sinkhorn, 50 iterations, implicit diag(r) P0 diag(c) ----
  #pragma unroll 1
  for (int it = 0; it < MAXIT; ++it) {
    // column normalize (reference axis=1): colsum[j] = c[j] * sum_i r[i] P0[i][j]
    for (int s = wv; s < 16; s += 8) col_combine(P, rV, colS, s * 16, lane);
    __syncthreads();
    {
      const float c  = cV[tid];
      const float cs = c * colS[tid];
      cV[tid] = c / fmaxf(cs, EPSV);
    }
    __syncthreads();
    // row normalize (axis=2): rowsum[i] = r[i] * sum_j P0[i][j] c[j]
    for (int s = wv; s < 16; s += 8) row_combine(P, cV, rowT, s * 16, lane);
    __syncthreads();
    {
      const float r  = rV[tid];
      const float rs = r * rowT[tid];
      rV[tid] = r / fmaxf(rs, EPSV);
    }
    __syncthreads();
  }

  // ---- phase 3: numerator = sum_i disc[i] * r[i] * sum_j P0[i][j]*c[j]*g[j]
  // (final pad_or->0 is implied: P0=0 on pad_or\pad_and, g=0 on masked cols)
  w2[tid] = cV[tid] * gn[tid];
  __syncthreads();
  for (int s = wv; s < 16; s += 8) row_combine(P, w2, rowT, s * 16, lane);
  __syncthreads();
  {
    const float disc = 1.0f / log2f((float)tid + 2.0f);
    red[tid] = disc * rV[tid] * rowT[tid];
  }
  __syncthreads();
  for (int st = 128; st > 0; st >>= 1) {
    if (tid < st) red[tid] += red[tid + st];
    __syncthreads();
  }
  const float numer = red[0];
  __syncthreads();

  // ---- phase 4: IDCG via pairwise-ranking (stable descending sort) ----
  {
    const bool  mj   = (yt[tid] == -1.0f);
    const float keyj = mj ? NEG_INF : yt[tid];
    int rank = 0;
    for (int k = 0; k < NN; ++k) {
      const float ytk  = yt[k];
      const float keyk = (ytk == -1.0f) ? NEG_INF : ytk;
      rank += (int)((keyk > keyj) || ((keyk == keyj) && (k < tid)));
    }
    red[tid] = gn[tid] / log2f((float)rank + 2.0f);
  }
  __syncthreads();
  for (int st = 128; st > 0; st >>= 1) {
    if (tid < st) red[tid] += red[tid + st];
    __syncthreads();
  }
  if (tid == 0) {
    const float idcg = red[0];
    const bool  zero = (idcg == 0.0f);
    ndcg_out[b] = zero ? 0.0f : numer / (idcg + EPSV);
    flag_out[b] = zero ? 0.0f : 1.0f;
  }
}

extern "C" __global__ void neural_ndcg_finish(const float* __restrict__ ndcg_in,
                                              const float* __restrict__ flag_in,
                                              float* __restrict__ out, int B) {
  __shared__ float ssum[128];
  __shared__ float scnt[128];
  const int t = threadIdx.x;
  float s = 0.f, c = 0.f;
  for (int i = t; i < B; i += 128) { s += ndcg_in[i]; c += flag_in[i]; }
  ssum[t] = s; scnt[t] = c;
  __syncthreads();
  for (int st = 64; st > 0; st >>= 1) {
    if (t < st) { ssum[t] += ssum[t + st]; scnt[t] += scnt[t + st]; }
    __syncthreads();
  }
  if (t == 0) {
    const float cnt = scnt[0];
    out[0] = (cnt == 0.f) ? 0.f : -(ssum[0] / cnt);
  }
}

extern "C" void kernel_launch(void* const* d_in, const int* in_sizes, int n_in,
                              void* d_out, int out_size, void* d_ws, size_t ws_size,
                              hipStream_t stream) {
  (void)n_in; (void)out_size; (void)ws_size;
  const float* y_pred = (const float*)d_in[0];
  const float* y_true = (const float*)d_in[1];
  const int B = in_sizes[0] / NN;
  if (B <= 0) return;

  float* ndcg_ws = (float*)d_ws;
  float* flag_ws = ndcg_ws + B;

  const size_t smem_bytes = (size_t)(NN * SROW + 11 * NN) * sizeof(float);
  hipFuncSetAttribute(reinterpret_cast<const void*>(neural_ndcg_main),
                      hipFuncAttributeMaxDynamicSharedMemorySize,
                      (int)smem_bytes);

  neural_ndcg_main<<<B, 256, smem_bytes, stream>>>(y_pred, y_true, ndcg_ws, flag_ws);
  neural_ndcg_finish<<<1, 128, 0, stream>>>(ndcg_ws, flag_ws, (float*)d_out, B);
}